// Gene_23003844838035
// MI455X (gfx1250) — compile-verified
//
#include <hip/hip_runtime.h>

#define NN     8192
#define HID    128
#define OUTD   64
#define NE     262144
#define BN_EPS 1e-5f

typedef __attribute__((ext_vector_type(2))) float v2f;
typedef __attribute__((ext_vector_type(8))) float v8f;

static __device__ __forceinline__ v8f wmma_f32(v2f a, v2f b, v8f c) {
  return __builtin_amdgcn_wmma_f32_16x16x4_f32(
      /*neg_a=*/false, a, /*neg_b=*/false, b,
      /*c_mod=*/(short)0, c, /*reuse_a=*/false, /*reuse_b=*/false);
}

// ---------------------------------------------------------------- utilities
__global__ void zero_f32(float* __restrict__ p, int n) {
  int i = blockIdx.x * blockDim.x + threadIdx.x;
  if (i < n) p[i] = 0.0f;
}

// ------------------------------------------------- segment_sum (scatter-add)
// one thread per (edge, feature): out[dst[e]][c] += feat[src[e]][c]
__global__ void edge_scatter(const float* __restrict__ feat,
                             const int* __restrict__ src,
                             const int* __restrict__ dst,
                             float* __restrict__ out) {
  int idx = blockIdx.x * blockDim.x + threadIdx.x;   // NE*HID = 2^25 threads
  int e = idx >> 7;          // / HID
  int c = idx & (HID - 1);   // % HID
  if (e < NE) {
    atomicAdd(&out[(size_t)dst[e] * HID + c], feat[(size_t)src[e] * HID + c]);
  }
}

// ------------------------------------------------- h2 = relu(ns1 @ w1 + b1)
__global__ void mlp_relu(const float* __restrict__ ns1,
                         const float* __restrict__ w1,
                         const float* __restrict__ w1b,
                         float* __restrict__ h2) {
  int idx = blockIdx.x * blockDim.x + threadIdx.x;   // NN*HID threads
  if (idx >= NN * HID) return;
  int n = idx >> 7;
  int c = idx & (HID - 1);
  const float* row = ns1 + (size_t)n * HID;
  float acc = w1b[c];
#pragma unroll 8
  for (int k = 0; k < HID; ++k) acc += row[k] * w1[k * HID + c];
  h2[idx] = fmaxf(acc, 0.0f);
}

// ---------------- h = (1-eps)*(x@fc_w + fc_b) + eps*(ns2@w2 + w2_b)  (fused)
__global__ void combine_h(const float* __restrict__ x,
                          const float* __restrict__ fc_w,
                          const float* __restrict__ fc_b,
                          const float* __restrict__ ns2,
                          const float* __restrict__ w2,
                          const float* __restrict__ w2_b,
                          const float* __restrict__ eps,
                          float* __restrict__ h) {
  int idx = blockIdx.x * blockDim.x + threadIdx.x;   // NN*OUTD threads
  if (idx >= NN * OUTD) return;
  int n = idx >> 6;
  int o = idx & (OUTD - 1);
  const float* xrow  = x   + (size_t)n * HID;
  const float* nrow  = ns2 + (size_t)n * HID;
  float h1 = fc_b[o];
  float h2 = w2_b[o];
#pragma unroll 8
  for (int k = 0; k < HID; ++k) {
    h1 += xrow[k] * fc_w[k * OUTD + o];
    h2 += nrow[k] * w2[k * OUTD + o];
  }
  float e = eps[n];
  h[idx] = (1.0f - e) * h1 + e * h2;
}

// ------------------- ret = 0.5*(h @ h^T + x @ x^T)  via V_WMMA_F32_16X16X4_F32
// One wave per 32x32 output block: 2x2 register-blocked 16x16 WMMA tiles.
// Per k-step: 2 A-fragments + 2 B-fragments feed 4 WMMAs (1.0 loads/WMMA,
// vs 2.0 unblocked) -> halves L2 fragment traffic behind the 268 MB store
// stream. Fragment addressing (f32 16x4 A / 4x16 B): lane L reads
// row (L&15) of its 16-row block, K sub-offset 2*(L>>4); C/D layout:
// lane L, vgpr v -> row v + 8*(L>>4), col L&15.
__global__ void __launch_bounds__(128)
gram_wmma(const float* __restrict__ x,
          const float* __restrict__ h,
          float* __restrict__ ret) {
  const int lane = threadIdx.x;                    // 0..31 (wave32)
  const int i2   = blockIdx.y;                     // 32-row block: 0..255
  const int j2   = blockIdx.x * 4 + threadIdx.y;   // 32-col block: 0..255

  const int r  = lane & 15;        // row within 16-row sub-block
  const int kh = (lane >> 4) * 2;  // K sub-offset: lanes 0-15 -> +0 ; 16-31 -> +2

  v8f c00 = {}, c01 = {}, c10 = {}, c11 = {};

  // ---- x @ x^T contribution, K = 128
  {
    const float* a0 = x + (size_t)(i2 * 32 + r) * HID + kh;
    const float* a1 = a0 + (size_t)16 * HID;
    const float* b0 = x + (size_t)(j2 * 32 + r) * HID + kh;
    const float* b1 = b0 + (size_t)16 * HID;
#pragma unroll
    for (int k0 = 0; k0 < HID; k0 += 4) {
      v2f av0 = *(const v2f*)(a0 + k0);
      v2f av1 = *(const v2f*)(a1 + k0);
      v2f bv0 = *(const v2f*)(b0 + k0);
      v2f bv1 = *(const v2f*)(b1 + k0);
      c00 = wmma_f32(av0, bv0, c00);
      c01 = wmma_f32(av0, bv1, c01);
      c10 = wmma_f32(av1, bv0, c10);
      c11 = wmma_f32(av1, bv1, c11);
    }
  }
  // ---- h @ h^T contribution, K = 64
  {
    const float* a0 = h + (size_t)(i2 * 32 + r) * OUTD + kh;
    const float* a1 = a0 + (size_t)16 * OUTD;
    const float* b0 = h + (size_t)(j2 * 32 + r) * OUTD + kh;
    const float* b1 = b0 + (size_t)16 * OUTD;
#pragma unroll
    for (int k0 = 0; k0 < OUTD; k0 += 4) {
      v2f av0 = *(const v2f*)(a0 + k0);
      v2f av1 = *(const v2f*)(a1 + k0);
      v2f bv0 = *(const v2f*)(b0 + k0);
      v2f bv1 = *(const v2f*)(b1 + k0);
      c00 = wmma_f32(av0, bv0, c00);
      c01 = wmma_f32(av0, bv1, c01);
      c10 = wmma_f32(av1, bv0, c10);
      c11 = wmma_f32(av1, bv1, c11);
    }
  }

  // ---- scale by 0.5 and store the 4 tiles (coalesced 16-lane row segments)
  const int col0 = j2 * 32 + (lane & 15);
  const int row0 = i2 * 32 + (lane >> 4) * 8;
#pragma unroll
  for (int v = 0; v < 8; ++v) {
    float* p0 = ret + (size_t)(row0 + v) * NN + col0;
    float* p1 = ret + (size_t)(row0 + v + 16) * NN + col0;
    p0[0]  = c00[v] * 0.5f;
    p0[16] = c01[v] * 0.5f;
    p1[0]  = c10[v] * 0.5f;
    p1[16] = c11[v] * 0.5f;
  }
}

// ---------------------------------- batch-norm over axis 0 (one block / col)
__global__ void bn_kernel(const float* __restrict__ h,
                          const float* __restrict__ gamma,
                          const float* __restrict__ beta,
                          float* __restrict__ hbn) {
  const int c   = blockIdx.x;       // 0..63
  const int tid = threadIdx.x;      // 0..255
  __shared__ float ssum[256];
  __shared__ float ssq[256];
  __shared__ float s_mean, s_inv;

  float s = 0.0f, q = 0.0f;
  for (int n = tid; n < NN; n += 256) {
    float v = h[(size_t)n * OUTD + c];
    s += v;
    q += v * v;
  }
  ssum[tid] = s;
  ssq[tid]  = q;
  __syncthreads();
  for (int off = 128; off > 0; off >>= 1) {
    if (tid < off) {
      ssum[tid] += ssum[tid + off];
      ssq[tid]  += ssq[tid + off];
    }
    __syncthreads();
  }
  if (tid == 0) {
    float mean = ssum[0] / (float)NN;
    float var  = ssq[0] / (float)NN - mean * mean;
    s_mean = mean;
    s_inv  = rsqrtf(var + BN_EPS);
  }
  __syncthreads();
  const float mean = s_mean;
  const float scl  = s_inv * gamma[c];
  const float bta  = beta[c];
  for (int n = tid; n < NN; n += 256) {
    hbn[(size_t)n * OUTD + c] = (h[(size_t)n * OUTD + c] - mean) * scl + bta;
  }
}

// --------------------------------------------------------------------- launch
extern "C" void kernel_launch(void* const* d_in, const int* in_sizes, int n_in,
                              void* d_out, int out_size, void* d_ws, size_t ws_size,
                              hipStream_t stream) {
  const float* x     = (const float*)d_in[0];
  // d_in[1] = adj (unused by reference math)
  const int*   src   = (const int*)d_in[2];
  const int*   dst   = (const int*)d_in[3];
  const float* fc_w  = (const float*)d_in[4];
  const float* fc_b  = (const float*)d_in[5];
  const float* w1    = (const float*)d_in[6];
  const float* w1_b  = (const float*)d_in[7];
  const float* w2    = (const float*)d_in[8];
  const float* w2_b  = (const float*)d_in[9];
  const float* eps   = (const float*)d_in[10];
  const float* gamma = (const float*)d_in[11];
  const float* beta  = (const float*)d_in[12];

  float* ret = (float*)d_out;                    // 8192*8192
  float* hbn = ret + (size_t)NN * NN;            // 8192*64

  // workspace layout (floats)
  float* ns1 = (float*)d_ws;                     // 8192*128
  float* h2  = ns1 + (size_t)NN * HID;           // 8192*128
  float* ns2 = h2  + (size_t)NN * HID;           // 8192*128
  float* h   = ns2 + (size_t)NN * HID;           // 8192*64

  // 1) zero the scatter accumulators
  {
    int n = NN * HID;
    zero_f32<<<(n + 255) / 256, 256, 0, stream>>>(ns1, n);
    zero_f32<<<(n + 255) / 256, 256, 0, stream>>>(ns2, n);
  }
  // 2) ns1 = segment_sum(x[src], dst)
  {
    int total = NE * HID;
    edge_scatter<<<total / 256, 256, 0, stream>>>(x, src, dst, ns1);
  }
  // 3) h2 = relu(ns1 @ w1 + w1_b)
  {
    int total = NN * HID;
    mlp_relu<<<total / 256, 256, 0, stream>>>(ns1, w1, w1_b, h2);
  }
  // 4) ns2 = segment_sum(h2[src], dst)
  {
    int total = NE * HID;
    edge_scatter<<<total / 256, 256, 0, stream>>>(h2, src, dst, ns2);
  }
  // 5) h = (1-eps)*(x@fc_w+fc_b) + eps*(ns2@w2+w2_b)
  {
    int total = NN * OUTD;
    combine_h<<<total / 256, 256, 0, stream>>>(x, fc_w, fc_b, ns2, w2, w2_b, eps, h);
  }
  // 6) ret = 0.5*(h@h^T + x@x^T)   [dominant kernel, fp32 WMMA, 2x2 blocked]
  {
    dim3 blk(32, 4);                      // 4 waves, each owns one 32x32 block
    dim3 grd(NN / 32 / 4, NN / 32);       // (64, 256)
    gram_wmma<<<grd, blk, 0, stream>>>(x, h, ret);
  }
  // 7) h_bn
  {
    bn_kernel<<<OUTD, 256, 0, stream>>>(h, gamma, beta, hbn);
  }
}